// OVDEIMDecoder_59914793779616
// MI455X (gfx1250) — compile-verified
//
#include <hip/hip_runtime.h>
#include <hip/hip_bf16.h>
#include <stdint.h>

typedef __attribute__((ext_vector_type(2))) float v2f;
typedef __attribute__((ext_vector_type(8))) float v8f;

// ---------------- geometry constants (from reference config) ----------------
#define BS   32
#define QN   300
#define CC   256
#define NHh  8
#define HD   32
#define SUMP 16
#define VV   8500
#define MROWS (BS * QN)          // 9600
#define N_OFF 256                // NH*SUMP*2
#define N_ATT 128                // NH*SUMP
#define NTILES_M (MROWS / 16)    // 600
#define NTILES_N ((N_OFF + N_ATT) / 16)  // 24
#define LDA 260                  // LDS row stride (dwords); 260 % 64 == 4 -> conflict-free frags

// Async global->LDS copy of 16 bytes (CDNA5 GLOBAL_LOAD_ASYNC_TO_LDS_B128).
// lds_off: LDS byte offset (low 32 bits of the flat shared address).
__device__ __forceinline__ void async_copy_b128(unsigned lds_off, const float* gaddr) {
    asm volatile("global_load_async_to_lds_b128 %0, %1, off"
                 :: "v"(lds_off), "v"(gaddr)
                 : "memory");
}
__device__ __forceinline__ void wait_asynccnt0() {
    asm volatile("s_wait_asynccnt 0x0" ::: "memory");
}

// ============================================================================
// Kernel 1: fused projection GEMM  P = query @ [W_off | W_attn] + bias
//   fp32 WMMA 16x16x4, one wave per 16x16 tile; 8 waves/block share one
//   16-row A tile, staged into LDS via async b128 DMA.
//   A fragment (16x4 f32): lanes 0-15 hold rows M=0..15 with (K, K+1),
//                          lanes 16-31 hold rows M=0..15 with (K+2, K+3).
//   B fragment (4x16 f32): lanes 0-15 hold cols N=0..15 of rows (K, K+1),
//                          lanes 16-31 hold cols N=0..15 of rows (K+2, K+3).
//   C/D (16x16 f32, 8 VGPRs): VGPR r -> M = r (lanes 0-15) / r+8 (lanes 16-31),
//                             N = lane&15.
// ============================================================================
__global__ __launch_bounds__(256) void ovdeim_proj_gemm(
    const float* __restrict__ query,    // [9600, 256] row-major
    const float* __restrict__ W_off,    // [256, 256]
    const float* __restrict__ b_off,    // [256]
    const float* __restrict__ W_attn,   // [256, 128]
    const float* __restrict__ b_attn,   // [128]
    float* __restrict__ ws_off,         // [9600, 256]
    float* __restrict__ ws_attn)        // [9600, 128]
{
    __shared__ float lAs[16 * LDA];     // 16 rows x 256 cols, padded

    const int lane = threadIdx.x & 31;
    const int wid  = threadIdx.x >> 5;
    const int tile = blockIdx.x * 8 + wid;      // 0 .. 14399
    // NTILES_N (24) is a multiple of 8 -> tm is uniform across the block.
    const int tmB  = (blockIdx.x * 8) / NTILES_N;   // 0 .. 599
    const int tn   = tile % NTILES_N;               // 0 .. 23
    const int half = lane >> 4;                     // 0: K,K+1   1: K+2,K+3
    const int l15  = lane & 15;

    // ---- stage the 16x256 A tile into LDS with async b128 DMA -------------
    // 1024 b128 chunks (16 rows x 64 chunks), 4 per thread.
    #pragma unroll
    for (int i = 0; i < 4; ++i) {
        const int idx = threadIdx.x + i * 256;      // 0 .. 1023
        const int r   = idx >> 6;
        const int c4  = (idx & 63) * 4;             // float column (16B chunks)
        const float* g = query + (long)(tmB * 16 + r) * CC + c4;
        const unsigned lo = (unsigned)(uintptr_t)&lAs[r * LDA + c4]; // 16B aligned
        async_copy_b128(lo, g);
    }
    wait_asynccnt0();
    __syncthreads();

    // ---- per-wave B addressing --------------------------------------------
    const bool isOff = (tn < N_OFF / 16);
    const float* Wp  = isOff ? W_off : W_attn;
    const int  ldw   = isOff ? N_OFF : N_ATT;
    const int  col   = (isOff ? tn : (tn - N_OFF / 16)) * 16 + l15;

    const float* aLds = &lAs[l15 * LDA + half * 2]; // A[l15, k + half*2 + {0,1}]
    const float* bptr = Wp + (long)(half * 2) * ldw + col;

    v8f c = {0.f, 0.f, 0.f, 0.f, 0.f, 0.f, 0.f, 0.f};

    #pragma unroll 4
    for (int k = 0; k < CC; k += 4) {
        v2f a = *(const v2f*)(aLds + k);            // ds_load_b64, conflict-free
        v2f b;
        b.x = bptr[(long)(k + 0) * ldw];
        b.y = bptr[(long)(k + 1) * ldw];
        c = __builtin_amdgcn_wmma_f32_16x16x4_f32(
                /*neg_a=*/false, a, /*neg_b=*/false, b,
                /*c_mod=*/(short)0, c, /*reuse_a=*/false, /*reuse_b=*/false);
    }

    const float bias = (isOff ? b_off : b_attn)[col];
    float* outp = isOff ? ws_off : ws_attn;
    const int ldo = ldw;

    #pragma unroll
    for (int r = 0; r < 8; ++r) {
        const int m = tmB * 16 + r + half * 8;
        outp[(long)m * ldo + col] = c[r] + bias;
    }
}

// ============================================================================
// Kernel 2: softmax + multi-scale deformable bilinear sampling.
//   One wave32 per (b, q, h); lane = head channel d (HD == 32).
//   All sampling-location / weight math is wave-uniform; value gathers are
//   fully coalesced 128B wave loads (channels contiguous in memory).
// ============================================================================
__global__ __launch_bounds__(256) void ovdeim_deform_attn(
    const float* __restrict__ value,    // [32, 8500, 8, 32]
    const float* __restrict__ refp,     // [32, 300, 1, 4] (cx, cy, w, h)
    const float* __restrict__ off,      // [9600, 256] = [bs,Q,NH,16,2]
    const float* __restrict__ attn,     // [9600, 128] = [bs,Q,NH,16] (logits)
    float* __restrict__ out)            // [32, 300, 256]
{
    const int lane = threadIdx.x & 31;
    const int wid  = threadIdx.x >> 5;
    const int g    = blockIdx.x * 8 + wid;      // 0 .. 76799
    const int h    = g & (NHh - 1);
    const int bq   = g >> 3;                    // 0 .. 9599
    const int b    = bq / QN;

    const int Hs[4] = {80, 40, 20, 10};
    const int VS[4] = {0, 6400, 8000, 8400};    // per-level row offsets in value

    const float* rp = refp + (long)bq * 4;
    const float cx = rp[0], cy = rp[1];
    const float sw = rp[2] * (0.25f * 0.5f);    // nps * offset_scale folded in
    const float sh = rp[3] * (0.25f * 0.5f);

    const float* offp = off  + (long)bq * N_OFF + h * (SUMP * 2);
    const float* ap   = attn + (long)bq * N_ATT + h * SUMP;

    // softmax over the 16 points (wave-uniform)
    float logits[SUMP];
    float mx = -3.4e38f;
    #pragma unroll
    for (int p = 0; p < SUMP; ++p) { logits[p] = ap[p]; mx = fmaxf(mx, logits[p]); }
    float ssum = 0.f;
    #pragma unroll
    for (int p = 0; p < SUMP; ++p) { logits[p] = __expf(logits[p] - mx); ssum += logits[p]; }
    const float inv = 1.0f / ssum;

    // value[b, v, h, lane]  ->  vbase + v * (NH*HD)
    const float* vbase = value + ((long)b * VV * NHh + h) * HD + lane;

    float acc = 0.f;
    #pragma unroll
    for (int p = 0; p < SUMP; ++p) {
        const int lvl = p >> 2;
        const int Wl = Hs[lvl], Hl = Hs[lvl], vs = VS[lvl];
        const float w_p = logits[p] * inv;

        const v2f o = *(const v2f*)(offp + 2 * p);  // 8B-aligned (x, y) pair

        const float x = (cx + o.x * sw) * (float)Wl - 0.5f;
        const float y = (cy + o.y * sh) * (float)Hl - 0.5f;
        const float xf = floorf(x), yf = floorf(y);
        const int   x0 = (int)xf,  y0 = (int)yf;
        const float fx = x - xf,   fy = y - yf;

        const float w00 = (1.f - fx) * (1.f - fy);
        const float w10 = fx * (1.f - fy);
        const float w01 = (1.f - fx) * fy;
        const float w11 = fx * fy;

        const bool xin0 = (x0 >= 0)     && (x0 < Wl);
        const bool xin1 = (x0 + 1 >= 0) && (x0 + 1 < Wl);
        const bool yin0 = (y0 >= 0)     && (y0 < Hl);
        const bool yin1 = (y0 + 1 >= 0) && (y0 + 1 < Hl);

        float g00 = 0.f, g10 = 0.f, g01 = 0.f, g11 = 0.f;
        if (yin0) {                                 // uniform branch: no divergence
            const float* rowp = vbase + (long)(vs + y0 * Wl) * (NHh * HD);
            if (xin0) g00 = rowp[(long)x0 * (NHh * HD)];
            if (xin1) g10 = rowp[(long)(x0 + 1) * (NHh * HD)];
        }
        if (yin1) {
            const float* rowp = vbase + (long)(vs + (y0 + 1) * Wl) * (NHh * HD);
            if (xin0) g01 = rowp[(long)x0 * (NHh * HD)];
            if (xin1) g11 = rowp[(long)(x0 + 1) * (NHh * HD)];
        }
        acc = fmaf(w_p, g00 * w00 + g10 * w10 + g01 * w01 + g11 * w11, acc);
    }

    out[(long)bq * CC + h * HD + lane] = acc;
}

// ============================================================================
extern "C" void kernel_launch(void* const* d_in, const int* in_sizes, int n_in,
                              void* d_out, int out_size, void* d_ws, size_t ws_size,
                              hipStream_t stream) {
    (void)in_sizes; (void)n_in; (void)out_size; (void)ws_size;
    const float* query  = (const float*)d_in[0];
    const float* refp   = (const float*)d_in[1];
    const float* value  = (const float*)d_in[2];
    const float* W_off  = (const float*)d_in[3];
    const float* b_off  = (const float*)d_in[4];
    const float* W_attn = (const float*)d_in[5];
    const float* b_attn = (const float*)d_in[6];
    float* out = (float*)d_out;

    float* ws_off  = (float*)d_ws;                         // 9600*256 f32
    float* ws_attn = ws_off + (long)MROWS * N_OFF;         // 9600*128 f32

    // 600 * 24 = 14400 wave-tiles, 8 waves (256 threads) per block
    ovdeim_proj_gemm<<<(NTILES_M * NTILES_N) / 8, 256, 0, stream>>>(
        query, W_off, b_off, W_attn, b_attn, ws_off, ws_attn);

    // 32*300*8 = 76800 (b,q,h) waves, 8 per block
    ovdeim_deform_attn<<<(BS * QN * NHh) / 8, 256, 0, stream>>>(
        value, refp, ws_off, ws_attn, out);
}